// Upsample_48095043780748
// MI455X (gfx1250) — compile-verified
//
#include <hip/hip_runtime.h>
#include <math.h>

// ---------------------------------------------------------------------------
// Problem constants (match the JAX reference)
// ---------------------------------------------------------------------------
#define N_SUP    20000
#define N_COARSE 8000
#define C_IN     256
#define C_OUT    64
#define EPS_LN   1e-5f
#define EPS_W    1e-8f

typedef float v2f __attribute__((ext_vector_type(2)));
typedef float v8f __attribute__((ext_vector_type(8)));

__device__ __forceinline__ float wave_reduce_sum(float x) {
#pragma unroll
    for (int off = 16; off > 0; off >>= 1)
        x += __shfl_xor(x, off, 32);
    return x;
}

// ---------------------------------------------------------------------------
// LayerNorm + GEMM (CK -> 64) using fp32 WMMA 16x16x4.
// One wave (32 threads) per 16 output rows.  A (normalized activations) is
// staged in LDS with +4 padding so the column reads (stride CK+4) are
// bank-conflict-free.  Output tile 16x64 = 4 WMMA accumulators.
//
// Fragment layouts (CDNA5 ISA 7.12.2, 32-bit 16x4 / 4x16 / 16x16):
//   A: lane&15 = M,  lanes<16 hold K = kb+0,kb+1 ; lanes>=16 hold K = kb+2,kb+3
//   B: lane&15 = N,  same K-half split by lane>=16
//   C/D: lane&15 = N, VGPR v -> M = v + 8*(lane>=16)
// ---------------------------------------------------------------------------
template <int CK>
__global__ __launch_bounds__(32)
void ln_gemm_kernel(const float* __restrict__ X,     // [nrows, CK]
                    const float* __restrict__ gam,   // [CK]
                    const float* __restrict__ bet,   // [CK]
                    const float* __restrict__ W,     // [CK, 64]
                    const float* __restrict__ bias,  // [64]
                    float* __restrict__ Y)           // [nrows, 64]
{
    __shared__ float As[16][CK + 4];

    const int lane  = threadIdx.x;
    const int rbase = blockIdx.x * 16;

    // ---- LayerNorm 16 rows (wave-cooperative, one row at a time) ----
    for (int r = 0; r < 16; ++r) {
        const float* xr = X + (size_t)(rbase + r) * CK;
        float vals[CK / 32];
        float s = 0.f, ss = 0.f;
#pragma unroll
        for (int u = 0; u < CK / 32; ++u) {
            float v = xr[lane + 32 * u];
            vals[u] = v;
            s  += v;
            ss += v * v;
        }
        s  = wave_reduce_sum(s);
        ss = wave_reduce_sum(ss);
        const float mean = s * (1.f / (float)CK);
        const float var  = ss * (1.f / (float)CK) - mean * mean;
        const float rstd = rsqrtf(var + EPS_LN);
#pragma unroll
        for (int u = 0; u < CK / 32; ++u) {
            const int c = lane + 32 * u;
            As[r][c] = (vals[u] - mean) * rstd * gam[c] + bet[c];
        }
    }
    __syncthreads();

    // ---- GEMM: Y[16,64] = As[16,CK] @ W[CK,64]  via fp32 WMMA ----
    v8f acc0 = {}, acc1 = {}, acc2 = {}, acc3 = {};
    const int m  = lane & 15;
    const int kh = (lane >> 4) << 1;   // 0 for lanes 0-15, 2 for lanes 16-31

    for (int kb = 0; kb < CK; kb += 4) {
        v2f a;
        a.x = As[m][kb + kh];
        a.y = As[m][kb + kh + 1];
        const float* w0p = W + (size_t)(kb + kh) * 64;
        const float* w1p = w0p + 64;

        v2f b;
        b.x = w0p[m];        b.y = w1p[m];
        acc0 = __builtin_amdgcn_wmma_f32_16x16x4_f32(false, a, false, b,
                                                     (short)0, acc0, false, false);
        b.x = w0p[16 + m];   b.y = w1p[16 + m];
        acc1 = __builtin_amdgcn_wmma_f32_16x16x4_f32(false, a, false, b,
                                                     (short)0, acc1, false, false);
        b.x = w0p[32 + m];   b.y = w1p[32 + m];
        acc2 = __builtin_amdgcn_wmma_f32_16x16x4_f32(false, a, false, b,
                                                     (short)0, acc2, false, false);
        b.x = w0p[48 + m];   b.y = w1p[48 + m];
        acc3 = __builtin_amdgcn_wmma_f32_16x16x4_f32(false, a, false, b,
                                                     (short)0, acc3, false, false);
    }

    // ---- Epilogue: + bias, store per C/D layout ----
    const float bv0 = bias[m];
    const float bv1 = bias[16 + m];
    const float bv2 = bias[32 + m];
    const float bv3 = bias[48 + m];
    const int roff = (lane >> 4) * 8;
#pragma unroll
    for (int v = 0; v < 8; ++v) {
        float* yr = Y + (size_t)(rbase + roff + v) * 64;
        yr[m]      = acc0[v] + bv0;
        yr[16 + m] = acc1[v] + bv1;
        yr[32 + m] = acc2[v] + bv2;
        yr[48 + m] = acc3[v] + bv3;
    }
}

// ---------------------------------------------------------------------------
// Brute-force k=3 NN + inverse-distance interpolation, accumulated into out
// (which already holds h1).  Coarse points staged through LDS in tiles of
// 1000 float4 (x,y,z,|x|^2) so the hot loop does ONE ds_load_b128 per
// candidate (wave-uniform address -> LDS broadcast).  d2 uses the reference
// form |s|^2 + |x|^2 - 2 s.x; strict '<' keeps first-occurrence tie order
// like jax.lax.top_k.  Gather/accumulate epilogue is float4-vectorized.
// ---------------------------------------------------------------------------
#define KNN_TILE 1000

__device__ __forceinline__ void top3_insert(float d2, int gj,
                                            float& bd0, float& bd1, float& bd2,
                                            int& bi0, int& bi1, int& bi2) {
    if (d2 < bd2) {
        if (d2 < bd0) {
            bd2 = bd1; bi2 = bi1;
            bd1 = bd0; bi1 = bi0;
            bd0 = d2;  bi0 = gj;
        } else if (d2 < bd1) {
            bd2 = bd1; bi2 = bi1;
            bd1 = d2;  bi1 = gj;
        } else {
            bd2 = d2;  bi2 = gj;
        }
    }
}

__global__ __launch_bounds__(256)
void knn_interp_kernel(const float* __restrict__ sxyz,  // [N_SUP, 3]
                       const float* __restrict__ cxyz,  // [N_COARSE, 3]
                       const float* __restrict__ h2,    // [N_COARSE, 64]
                       float* __restrict__ out)         // [N_SUP, 64] (+=)
{
    __shared__ float4 tile[KNN_TILE];   // (x, y, z, |x|^2) : 16 KB

    const int i = blockIdx.x * 256 + threadIdx.x;
    const bool valid = (i < N_SUP);

    float sx = 0.f, sy = 0.f, sz = 0.f, sn = 0.f;
    if (valid) {
        sx = sxyz[3 * i + 0];
        sy = sxyz[3 * i + 1];
        sz = sxyz[3 * i + 2];
        sn = sx * sx + sy * sy + sz * sz;
    }

    float bd0 = INFINITY, bd1 = INFINITY, bd2 = INFINITY;
    int   bi0 = 0, bi1 = 0, bi2 = 0;

    for (int t0 = 0; t0 < N_COARSE; t0 += KNN_TILE) {
        __syncthreads();
        for (int j = threadIdx.x; j < KNN_TILE; j += 256) {
            const float x = cxyz[3 * (t0 + j) + 0];
            const float y = cxyz[3 * (t0 + j) + 1];
            const float z = cxyz[3 * (t0 + j) + 2];
            tile[j] = make_float4(x, y, z, x * x + y * y + z * z);
        }
        __syncthreads();

#pragma unroll 4
        for (int j = 0; j < KNN_TILE; ++j) {
            const float4 t = tile[j];
            const float d2 = sn + t.w
                           - 2.f * (sx * t.x + sy * t.y + sz * t.z);
            top3_insert(d2, t0 + j, bd0, bd1, bd2, bi0, bi1, bi2);
        }
    }

    if (valid) {
        const float d0 = sqrtf(fmaxf(bd0, 0.f));
        const float d1 = sqrtf(fmaxf(bd1, 0.f));
        const float d2 = sqrtf(fmaxf(bd2, 0.f));
        const float r0 = 1.f / (d0 + EPS_W);
        const float r1 = 1.f / (d1 + EPS_W);
        const float r2 = 1.f / (d2 + EPS_W);
        const float rs = r0 + r1 + r2;
        const float w0 = r0 / rs, w1 = r1 / rs, w2 = r2 / rs;

        // rows are 256-byte aligned -> float4 OK
        const float4* p0 = (const float4*)(h2 + (size_t)bi0 * 64);
        const float4* p1 = (const float4*)(h2 + (size_t)bi1 * 64);
        const float4* p2 = (const float4*)(h2 + (size_t)bi2 * 64);
        float4* orow = (float4*)(out + (size_t)i * 64);
#pragma unroll
        for (int c = 0; c < 16; ++c) {
            float4 o = orow[c];
            const float4 a = p0[c];
            const float4 b = p1[c];
            const float4 d = p2[c];
            o.x += w0 * a.x + w1 * b.x + w2 * d.x;
            o.y += w0 * a.y + w1 * b.y + w2 * d.y;
            o.z += w0 * a.z + w1 * b.z + w2 * d.z;
            o.w += w0 * a.w + w1 * b.w + w2 * d.w;
            orow[c] = o;
        }
    }
}

// ---------------------------------------------------------------------------
// Launch.  Input order follows setup_inputs():
//  0 support_xyz  1 support_feats  2 support_offset  3 xyz  4 feats  5 offset
//  6 ln1_g  7 ln1_b  8 w1  9 b1  10 ln2_g  11 ln2_b  12 w2  13 b2
// ---------------------------------------------------------------------------
extern "C" void kernel_launch(void* const* d_in, const int* in_sizes, int n_in,
                              void* d_out, int out_size, void* d_ws, size_t ws_size,
                              hipStream_t stream) {
    const float* support_xyz   = (const float*)d_in[0];
    const float* support_feats = (const float*)d_in[1];
    const float* xyz           = (const float*)d_in[3];
    const float* feats         = (const float*)d_in[4];
    const float* ln1_g         = (const float*)d_in[6];
    const float* ln1_b         = (const float*)d_in[7];
    const float* w1            = (const float*)d_in[8];
    const float* b1            = (const float*)d_in[9];
    const float* ln2_g         = (const float*)d_in[10];
    const float* ln2_b         = (const float*)d_in[11];
    const float* w2            = (const float*)d_in[12];
    const float* b2            = (const float*)d_in[13];

    float* out = (float*)d_out;
    float* h2  = (float*)d_ws;   // [N_COARSE, 64] = 2 MB scratch

    // h1 -> d_out (full overwrite every call; graph-replay safe)
    ln_gemm_kernel<C_OUT><<<N_SUP / 16, 32, 0, stream>>>(
        support_feats, ln1_g, ln1_b, w1, b1, out);

    // h2 -> workspace
    ln_gemm_kernel<C_IN><<<N_COARSE / 16, 32, 0, stream>>>(
        feats, ln2_g, ln2_b, w2, b2, h2);

    // out += interpolation(h2)
    knn_interp_kernel<<<(N_SUP + 255) / 256, 256, 0, stream>>>(
        support_xyz, xyz, h2, out);
}